// Sampler_74929999446258
// MI455X (gfx1250) — compile-verified
//
#include <hip/hip_runtime.h>
#include <stdint.h>

// ---------------- problem constants (from setup_inputs) ----------------
#define NPTS   20000          // points per cloud
#define MSEL   2000           // TARGETED_NUM_POINTS
#define BATCH  8
#define BLOCK  1024           // 32 wave32 waves -> one full WGP
#define NWAVES (BLOCK / 32)   // 32
#define KMAX   ((NPTS + BLOCK - 1) / BLOCK)   // 20 dist entries per thread (registers)

typedef __attribute__((address_space(1))) int as1_int;
typedef __attribute__((address_space(3))) int as3_int;

// ---- CDNA5 async global->LDS copy (per-lane gather/scatter transpose) ----
__device__ __forceinline__ void async_copy_b32(const void* gsrc, void* ldst) {
#if defined(__HIP_DEVICE_COMPILE__)
# if __has_builtin(__builtin_amdgcn_global_load_async_to_lds_b32)
  __builtin_amdgcn_global_load_async_to_lds_b32((as1_int*)gsrc, (as3_int*)ldst, 0, 0);
# else
  unsigned           loff = (unsigned)(unsigned long long)(as3_int*)ldst;
  unsigned long long ga   = (unsigned long long)gsrc;
  asm volatile("global_load_async_to_lds_b32 %0, %1, off"
               :: "v"(loff), "v"(ga) : "memory");
# endif
#else
  (void)gsrc; (void)ldst;
#endif
}

__device__ __forceinline__ void async_wait0() {
#if defined(__HIP_DEVICE_COMPILE__)
# if __has_builtin(__builtin_amdgcn_s_wait_asynccnt)
  __builtin_amdgcn_s_wait_asynccnt(0);
# else
  asm volatile("s_wait_asynccnt 0" ::: "memory");
# endif
#endif
}

// ------------------------- FPS + gather kernel -------------------------
// One workgroup per cloud. Coordinates live in LDS (SoA: float2 xy + float z),
// running min-distance lives in VGPRs (KMAX floats/thread).
__global__ __launch_bounds__(BLOCK) void fps_sample_kernel(
    const float* __restrict__ C,   // [BATCH*NPTS*3]
    const float* __restrict__ F,   // [BATCH*NPTS*nfeat]
    float* __restrict__ outC,      // [BATCH*MSEL*3]
    float* __restrict__ outF,      // [BATCH*MSEL*nfeat]
    int nfeat)
{
  __shared__ float2 s_xy[NPTS];        // 160 000 B
  __shared__ float  s_z [NPTS];        //  80 000 B
  __shared__ int    s_sel[MSEL];       //   8 000 B
  __shared__ float  s_redv[NWAVES];
  __shared__ int    s_redi[NWAVES];
  __shared__ float  s_bc[3];           // broadcast coords of last pick

  const int b   = blockIdx.x;
  const int tid = threadIdx.x;
  const float* Cb = C + (size_t)b * NPTS * 3;

  // ---- Stage coords: global AoS -> LDS SoA via async-to-LDS gather ----
  for (int i = tid; i < NPTS; i += BLOCK) {
    const float* g = Cb + (size_t)i * 3;
    async_copy_b32(g + 0, &s_xy[i].x);
    async_copy_b32(g + 1, &s_xy[i].y);
    async_copy_b32(g + 2, &s_z[i]);
  }
  async_wait0();
  if (tid == 0) s_sel[0] = 0;          // reference: first index is 0
  __syncthreads();

  // ---- running min-distance in registers ----
  float dist[KMAX];
#pragma unroll
  for (int k = 0; k < KMAX; ++k) dist[k] = 1e10f;

  float lx = s_xy[0].x, ly = s_xy[0].y, lz = s_z[0];

  for (int it = 1; it < MSEL; ++it) {
    float bestv = -1.0f;               // dists are >= 0
    int   besti = 0x7FFFFFFF;
#pragma unroll
    for (int k = 0; k < KMAX; ++k) {
      const int i = tid + k * BLOCK;
      if (i < NPTS) {
        const float2 xy = s_xy[i];
        const float dx = xy.x - lx;
        const float dy = xy.y - ly;
        const float dz = s_z[i] - lz;
        const float d  = dx * dx + dy * dy + dz * dz;
        const float dm = fminf(dist[k], d);
        dist[k] = dm;
        if (dm > bestv) { bestv = dm; besti = i; }  // strict > => first max (smallest i per thread)
      }
    }
    // wave32 argmax reduce; ties -> smaller index (matches jnp.argmax)
#pragma unroll
    for (int off = 16; off > 0; off >>= 1) {
      const float ov = __shfl_xor(bestv, off, 32);
      const int   oi = __shfl_xor(besti, off, 32);
      if (ov > bestv || (ov == bestv && oi < besti)) { bestv = ov; besti = oi; }
    }
    if ((tid & 31) == 0) { s_redv[tid >> 5] = bestv; s_redi[tid >> 5] = besti; }
    __syncthreads();
    if (tid < 32) {                    // NWAVES == 32: one wave finishes it
      float v  = s_redv[tid];
      int   vi = s_redi[tid];
#pragma unroll
      for (int off = 16; off > 0; off >>= 1) {
        const float ov = __shfl_xor(v, off, 32);
        const int   oi = __shfl_xor(vi, off, 32);
        if (ov > v || (ov == v && oi < vi)) { v = ov; vi = oi; }
      }
      if (tid == 0) {
        s_sel[it] = vi;
        const float2 w = s_xy[vi];
        s_bc[0] = w.x; s_bc[1] = w.y; s_bc[2] = s_z[vi];
      }
    }
    __syncthreads();
    lx = s_bc[0]; ly = s_bc[1]; lz = s_bc[2];
  }
  // s_sel is complete (barrier at end of last iteration)

  // ---- gather sampled coordinates from LDS ----
  float* oC = outC + (size_t)b * MSEL * 3;
  for (int j = tid; j < MSEL * 3; j += BLOCK) {
    const int row  = j / 3;
    const int comp = j - row * 3;
    const int src  = s_sel[row];
    oC[j] = (comp == 0) ? s_xy[src].x : (comp == 1) ? s_xy[src].y : s_z[src];
  }

  // ---- gather sampled features (float4 vectorized; nfeat % 4 == 0) ----
  const float* Fb = F    + (size_t)b * NPTS * nfeat;
  float*       oF = outF + (size_t)b * MSEL * nfeat;
  const int c4 = nfeat >> 2;
  for (int j = tid; j < MSEL * c4; j += BLOCK) {
    const int row = j / c4;
    const int col = j - row * c4;
    const float4* srow = (const float4*)(Fb + (size_t)s_sel[row] * nfeat);
    ((float4*)(oF + (size_t)row * nfeat))[col] = srow[col];
  }
}

// ------------------------------ launcher -------------------------------
extern "C" void kernel_launch(void* const* d_in, const int* in_sizes, int n_in,
                              void* d_out, int out_size, void* d_ws, size_t ws_size,
                              hipStream_t stream) {
  (void)n_in; (void)out_size; (void)d_ws; (void)ws_size;
  const float* C = (const float*)d_in[0];
  const float* F = (const float*)d_in[1];

  const int total_pts = in_sizes[0] / 3;            // BATCH * NPTS
  const int nfeat     = in_sizes[1] / total_pts;    // 128

  float* outC = (float*)d_out;                                  // [BATCH*MSEL*3]
  float* outF = outC + (size_t)BATCH * MSEL * 3;                // [BATCH*MSEL*nfeat]

  fps_sample_kernel<<<BATCH, BLOCK, 0, stream>>>(C, F, outC, outF, nfeat);
}